// DoReFaLike_Quantizer_82566451299165
// MI455X (gfx1250) — compile-verified
//
#include <hip/hip_runtime.h>
#include <math.h>

// DoReFa-like quantizer for x of shape (out_c, in_c, h, w) = (2048, 2048, 3, 3), f32.
//   q = round(tanh(x) / max|tanh(x)| * 127); per group of 8 in-channels keep the
//   4 largest-|q| (stable top-k); out = q / 127.
//
// Memory-bound: ~453 MB total traffic => ~19 us at 23.3 TB/s. To stay under that
// roofline the per-element VALU cost must be ~<20 slots: we use the gfx1250
// hardware v_tanh_f32 (TRANS op), precomputed reciprocals instead of IEEE
// divides, and a 28-comparison pairwise rank for the top-4 mask.
// Pass 1 uses max|tanh(x)| == tanh(max|x|) (tanh odd+monotone) so it is a pure
// |x| max-reduction. Pass 2 stages 72-float contiguous tiles through LDS with
// CDNA5 async global<->LDS b128 transfers (ASYNCcnt) so all global traffic is
// coalesced 128-bit even though the group axis is strided (9 floats).
//
// Assumes n % 2304 == 0 and in_c % 8 == 0 (true for the harness setup).

#define GROUP 8
#define KEEP 4
#define HW 9
#define TILE_FLOATS 72                               // 8 channels * 9 spatial, contiguous
#define TILES_PER_BLOCK 32
#define BLOCK_THREADS (TILES_PER_BLOCK * HW)         // 288 threads = 9 waves
#define LDS_FLOATS (TILES_PER_BLOCK * TILE_FLOATS)   // 2304 floats = 9216 B

#if defined(__has_builtin)
#if __has_builtin(__builtin_amdgcn_global_load_async_to_lds_b128) && \
    __has_builtin(__builtin_amdgcn_global_store_async_from_lds_b128)
#define HAVE_ASYNC_LDS 1
#endif
#endif
#ifndef HAVE_ASYNC_LDS
#define HAVE_ASYNC_LDS 0
#endif

#if HAVE_ASYNC_LDS
// Builtin params are pointers to 4 x i32 vectors in AS1 (global) / AS3 (LDS).
typedef int v4i __attribute__((ext_vector_type(4)));
typedef v4i __attribute__((address_space(1)))* gptr_v4i;
typedef v4i __attribute__((address_space(3)))* lptr_v4i;
#endif

__device__ __forceinline__ void wait_async0() {
#if defined(__has_builtin) && __has_builtin(__builtin_amdgcn_s_wait_asynccnt)
  __builtin_amdgcn_s_wait_asynccnt(0);
#else
  asm volatile("s_wait_asynccnt 0" ::: "memory");
#endif
}

// gfx1250 has a hardware TANH transcendental (1 TRANS issue vs ~30 VALU for
// the OCML expansion). Same accuracy class as XLA's own tanh approximation.
__device__ __forceinline__ float fast_tanh(float x) {
#if defined(__has_builtin) && __has_builtin(__builtin_amdgcn_tanhf)
  return __builtin_amdgcn_tanhf(x);
#elif defined(__has_builtin) && __has_builtin(__builtin_amdgcn_tanh_f32)
  return __builtin_amdgcn_tanh_f32(x);
#else
  return tanhf(x);
#endif
}

// ---------------- Pass 2: quantize + per-group top-4 mask -------------------
// (defined first so the disasm snippet shows this kernel's body)
__global__ void __launch_bounds__(BLOCK_THREADS) quant_kernel(
    const float* __restrict__ x, float* __restrict__ out,
    const float* __restrict__ wsf) {
  __shared__ float lds[LDS_FLOATS];
  const int tid = threadIdx.x;
  const long base = (long)blockIdx.x * LDS_FLOATS;

  // ---- stage 9216 B of contiguous input into LDS (2 x async b128/thread) ----
#if HAVE_ASYNC_LDS
  {
    const float* g0 = x + base + (long)tid * 4;
    const float* g1 = g0 + BLOCK_THREADS * 4;
    __builtin_amdgcn_global_load_async_to_lds_b128(
        (gptr_v4i)(v4i*)g0, (lptr_v4i)(v4i*)&lds[tid * 4], 0, 0);
    __builtin_amdgcn_global_load_async_to_lds_b128(
        (gptr_v4i)(v4i*)g1, (lptr_v4i)(v4i*)&lds[(tid + BLOCK_THREADS) * 4], 0, 0);
    wait_async0();  // this wave's async loads landed in LDS
  }
#else
  {
    const float4* g4 = (const float4*)(x + base);
    float4* l4 = (float4*)lds;
    l4[tid] = g4[tid];
    l4[tid + BLOCK_THREADS] = g4[tid + BLOCK_THREADS];
  }
#endif
  __syncthreads();  // all waves' tiles visible

  const float scale = wsf[1];      // delta / max|tanh(x)|
  const float inv_delta = wsf[2];  // 1 / delta

  // thread -> one (tile, hw) group; its 8 members sit at stride 9 in LDS and
  // are touched by no other thread, so in-place update needs no extra barrier.
  const int tile = tid / HW;
  const int hw = tid - tile * HW;
  const int lbase = tile * TILE_FLOATS + hw;

  float q[GROUP], a[GROUP];
#pragma unroll
  for (int j = 0; j < GROUP; ++j) {
    float t = fast_tanh(lds[lbase + HW * j]);
    q[j] = rintf(t * scale);  // round-half-even == jnp.round
    a[j] = fabsf(q[j]);
  }
  // Stable top-KEEP by |q| (ties -> lower index), exactly like jax.lax.top_k.
  // One comparison per pair feeds both ranks: for i<j, "i beats j" iff
  // a[i] >= a[j] (tie -> lower index wins); otherwise j beats i.
  int rank[GROUP];
#pragma unroll
  for (int i = 0; i < GROUP; ++i) rank[i] = 0;
#pragma unroll
  for (int i = 0; i < GROUP; ++i) {
#pragma unroll
    for (int j = i + 1; j < GROUP; ++j) {
      const int c = (a[i] >= a[j]);
      rank[j] += c;
      rank[i] += 1 - c;
    }
  }
#pragma unroll
  for (int j = 0; j < GROUP; ++j) {
    float v = (rank[j] < KEEP) ? q[j] : 0.0f;
    lds[lbase + HW * j] = v * inv_delta;
  }
  __syncthreads();  // results in LDS complete before bulk store

  // ---- coalesced async b128 store of the whole block's results ----
#if HAVE_ASYNC_LDS
  {
    float* g0 = out + base + (long)tid * 4;
    float* g1 = g0 + BLOCK_THREADS * 4;
    __builtin_amdgcn_global_store_async_from_lds_b128(
        (gptr_v4i)(v4i*)g0, (lptr_v4i)(v4i*)&lds[tid * 4], 0, 0);
    __builtin_amdgcn_global_store_async_from_lds_b128(
        (gptr_v4i)(v4i*)g1, (lptr_v4i)(v4i*)&lds[(tid + BLOCK_THREADS) * 4], 0, 0);
    wait_async0();  // drain before LDS dealloc at wave end
  }
#else
  {
    const float4* l4 = (const float4*)lds;
    float4* g4 = (float4*)(out + base);
    g4[tid] = l4[tid];
    g4[tid + BLOCK_THREADS] = l4[tid + BLOCK_THREADS];
  }
#endif
}

// ---------------- Pass 0: init workspace (harness poisons ws with 0xAA) ----
__global__ void init_ws_kernel(unsigned* ws) {
  if (threadIdx.x == 0) ws[0] = 0u;
}

// ---------------- Pass 1: global max|x| (float4 grid-stride + atomicMax) ----
__global__ void __launch_bounds__(256) absmax_kernel(const float4* __restrict__ x4,
                                                     int n4,
                                                     unsigned* __restrict__ wsmax) {
  float m = 0.0f;
  for (int i = blockIdx.x * blockDim.x + threadIdx.x; i < n4;
       i += gridDim.x * blockDim.x) {
    float4 v = x4[i];
    m = fmaxf(m, fmaxf(fmaxf(fabsf(v.x), fabsf(v.y)),
                       fmaxf(fabsf(v.z), fabsf(v.w))));
  }
  __shared__ float red[256];
  red[threadIdx.x] = m;
  __syncthreads();
  for (int s = 128; s > 0; s >>= 1) {
    if ((int)threadIdx.x < s)
      red[threadIdx.x] = fmaxf(red[threadIdx.x], red[threadIdx.x + s]);
    __syncthreads();
  }
  // |x| >= 0 so the uint bit pattern orders identically to the float value.
  if (threadIdx.x == 0) atomicMax(wsmax, __float_as_uint(red[0]));
}

// ------- Pass 1b: scalar finalize: scale = delta/tanh(max|x|), 1/delta ------
__global__ void finalize_kernel(const unsigned* __restrict__ wsmax,
                                const int* __restrict__ bits,
                                float* __restrict__ wsf) {
  const float maxabs = __uint_as_float(wsmax[0]);
  const float maxt = fast_tanh(maxabs);  // == max |tanh(x)| (tanh monotone, odd)
  const float delta = (float)((1 << (bits[0] - 1)) - 1);
  wsf[1] = delta / maxt;
  wsf[2] = 1.0f / delta;
}

extern "C" void kernel_launch(void* const* d_in, const int* in_sizes, int n_in,
                              void* d_out, int out_size, void* d_ws, size_t ws_size,
                              hipStream_t stream) {
  const float* x = (const float*)d_in[0];
  const int* bits = (const int*)d_in[1];
  float* out = (float*)d_out;
  const int n = in_sizes[0];  // 2048*2048*3*3 = 37,748,736

  unsigned* wsu = (unsigned*)d_ws;
  float* wsf = (float*)d_ws;

  init_ws_kernel<<<1, 32, 0, stream>>>(wsu);

  const int n4 = n >> 2;
  absmax_kernel<<<4096, 256, 0, stream>>>((const float4*)x, n4, wsu);

  finalize_kernel<<<1, 1, 0, stream>>>(wsu, bits, wsf);

  const int nblocks = n / LDS_FLOATS;  // 16384 for the harness shape
  quant_kernel<<<nblocks, BLOCK_THREADS, 0, stream>>>(x, out, wsf);
}